// PoolReduce_25503515803836
// MI455X (gfx1250) — compile-verified
//
#include <hip/hip_runtime.h>

// Vector types that clang's __builtin_nontemporal_load accepts cleanly.
typedef int   vi4 __attribute__((ext_vector_type(4)));
typedef float vf4 __attribute__((ext_vector_type(4)));

// Non-returning device-scope FP32 atomic add, SADDR (GVS) form:
//   global_atomic_add_f32 v_byteoff, v_data, s[base:base+1] scope:SCOPE_DEV
// RMW resolves in L2 (device coherence point); tracked by STOREcnt, covered
// by S_ENDPGM's implicit wait-idle.
__device__ __forceinline__ void atomic_add_f32_dev(float* base, int byte_off, float val) {
    asm volatile("global_atomic_add_f32 %0, %1, %2 scope:SCOPE_DEV"
                 :
                 : "v"(byte_off), "v"(val), "s"(base)
                 : "memory");
}

__global__ void zero_out_kernel(float* __restrict__ out, int n) {
    int i = blockIdx.x * blockDim.x + threadIdx.x;
    if (i < n) out[i] = 0.0f;
}

__global__ void __launch_bounds__(256)
scatter_add_kernel(const int* __restrict__ col,
                   const float* __restrict__ val,
                   float* __restrict__ out,
                   int nvec, int n) {
    const int stride = (int)(gridDim.x * blockDim.x);
    const int gid    = (int)(blockIdx.x * blockDim.x + threadIdx.x);

    const vi4* __restrict__ cv = reinterpret_cast<const vi4*>(col);
    const vf4* __restrict__ vv = reinterpret_cast<const vf4*>(val);

    // Main loop: two 128-bit non-temporal streaming loads (th:TH_LOAD_NT keeps
    // the 160MB one-pass stream from evicting the 400KB accumulator in L2),
    // then 4 non-returning scoped atomics. All addressing is 32-bit
    // (saddr + voffset), so per-atomic overhead is one v_lshlrev_b32.
    for (int i = gid; i < nvec; i += stride) {
        vi4 c = __builtin_nontemporal_load(cv + i);
        vf4 v = __builtin_nontemporal_load(vv + i);
        atomic_add_f32_dev(out, c.x << 2, v.x);
        atomic_add_f32_dev(out, c.y << 2, v.y);
        atomic_add_f32_dev(out, c.z << 2, v.z);
        atomic_add_f32_dev(out, c.w << 2, v.w);
    }

    // Scalar tail (n not a multiple of 4).
    for (int i = nvec * 4 + gid; i < n; i += stride) {
        atomic_add_f32_dev(out, col[i] << 2, val[i]);
    }
}

extern "C" void kernel_launch(void* const* d_in, const int* in_sizes, int n_in,
                              void* d_out, int out_size, void* d_ws, size_t ws_size,
                              hipStream_t stream) {
    // setup_inputs order: row_idx(0, unused), col_idx(1), values(2),
    //                     n_rows(3, unused), n_cols(4, == out_size)
    const int*   col = reinterpret_cast<const int*>(d_in[1]);
    const float* val = reinterpret_cast<const float*>(d_in[2]);
    float*       out = reinterpret_cast<float*>(d_out);

    const int n    = in_sizes[2];  // NNZ
    const int nvec = n / 4;

    // 1) Zero the accumulator (d_out is poisoned by the harness).
    {
        const int threads = 256;
        const int blocks  = (out_size + threads - 1) / threads;
        zero_out_kernel<<<blocks, threads, 0, stream>>>(out, out_size);
    }

    // 2) Scatter-add. 4096 blocks x 256 threads (8 wave32 per block) saturates
    //    all WGPs; each thread runs ~5 vec4 iterations.
    {
        const int threads = 256;
        long long want    = ((long long)nvec + threads - 1) / threads;
        int blocks        = (int)((want < 4096) ? (want > 0 ? want : 1) : 4096);
        scatter_add_kernel<<<blocks, threads, 0, stream>>>(col, val, out, nvec, n);
    }
}